// ExpertChoiceRouter_80857054314540
// MI455X (gfx1250) — compile-verified
//
#include <hip/hip_runtime.h>
#include <hip/hip_bf16.h>

typedef __attribute__((ext_vector_type(2))) float v2f;
typedef __attribute__((ext_vector_type(8))) float v8f;

#define HIDDEN 1024
#define NUM_EXPERTS 8
#define BATCH 4
#define SEQ 4096
#define CAPACITY 640   // 4096/8 * 1.25

// -----------------------------------------------------------------------------
// Kernel 1: fused router logits (WMMA f32 16x16x4) + softmax over experts.
// Each wave handles 16 tokens x 16 experts (experts 8..15 are zero padding).
// Block = 256 threads = 8 waves = 128 tokens. Grid = 16384/128 = 128 blocks.
// Writes softmax weights transposed to ws as (B, E, S) for the top-k pass.
// -----------------------------------------------------------------------------
__global__ __launch_bounds__(256)
void router_softmax_kernel(const float* __restrict__ hidden,
                           const float* __restrict__ W,
                           const float* __restrict__ bias,
                           float* __restrict__ wsW) {
    // Zero-padded expert weights in LDS: 16 rows x 1024 = 64KB.
    __shared__ float sW[16 * HIDDEN];
    for (int i = threadIdx.x; i < 16 * HIDDEN; i += blockDim.x) {
        int e = i >> 10;           // i / 1024
        int k = i & (HIDDEN - 1);
        sW[i] = (e < NUM_EXPERTS) ? W[e * HIDDEN + k] : 0.0f;
    }
    __syncthreads();

    const int lane  = threadIdx.x & 31;
    const int wave  = threadIdx.x >> 5;
    const int tokBase = blockIdx.x * 128 + wave * 16;   // 16 tokens per wave
    const int m    = lane & 15;        // token-within-tile for A, expert for B/C
    const int half = lane >> 4;        // lane half selects K sub-pair / token half

    // A fragment: lane holds hidden[tokBase + m][k0 + 2*half + {0,1}]
    const float* aRow = hidden + (size_t)(tokBase + m) * HIDDEN + 2 * half;
    // B fragment: lane holds sW[m][k0 + 2*half + {0,1}]  (m = expert column)
    const float* bRow = sW + m * HIDDEN + 2 * half;

    v8f c = {};
#pragma unroll 8
    for (int k0 = 0; k0 < HIDDEN; k0 += 4) {
        v2f a = *(const v2f*)(aRow + k0);
        v2f b = *(const v2f*)(bRow + k0);
        c = __builtin_amdgcn_wmma_f32_16x16x4_f32(
            /*neg_a=*/false, a, /*neg_b=*/false, b,
            /*c_mod=*/(short)0, c, /*reuse_a=*/false, /*reuse_b=*/false);
    }

    // Bias: lane's expert column is m; padded experts keep logit 0 but are
    // never reduced with the real ones (shfl_xor masks 1/2/4 stay in groups
    // of 8 lanes: {0-7} and {16-23} hold the 8 real experts).
    const float bv = (m < NUM_EXPERTS) ? bias[m] : 0.0f;

    float w[8];
#pragma unroll
    for (int j = 0; j < 8; ++j) {
        float v = c[j] + bv;
        float mx = v;
        mx = fmaxf(mx, __shfl_xor(mx, 1, 32));
        mx = fmaxf(mx, __shfl_xor(mx, 2, 32));
        mx = fmaxf(mx, __shfl_xor(mx, 4, 32));
        float ex = __expf(v - mx);
        float sm = ex;
        sm += __shfl_xor(sm, 1, 32);
        sm += __shfl_xor(sm, 2, 32);
        sm += __shfl_xor(sm, 4, 32);
        w[j] = ex / sm;
    }

    // Store to ws as (B, E, S). c[j] is token M = j + 8*half, expert m.
    if (m < NUM_EXPERTS) {
        const int bIdx = tokBase / SEQ;
        const int sBase = (tokBase & (SEQ - 1)) + 8 * half;   // 8-aligned
        float* dst = wsW + ((size_t)(bIdx * NUM_EXPERTS + m)) * SEQ + sBase;
        float4 lo = make_float4(w[0], w[1], w[2], w[3]);
        float4 hi = make_float4(w[4], w[5], w[6], w[7]);
        *(float4*)(dst)     = lo;
        *(float4*)(dst + 4) = hi;
    }
}

// -----------------------------------------------------------------------------
// Kernel 2: per-(b,e) top-640 of 4096 via full bitonic sort in LDS.
// Key = (float_bits << 32) | (0xFFFFFFFF - idx): descending sort gives
// value-descending with index-ascending tie break (weights are positive).
// Grid = 32 blocks (B*E), 512 threads, 32KB LDS.
// -----------------------------------------------------------------------------
__global__ __launch_bounds__(512)
void topk_sort_kernel(const float* __restrict__ wsW,
                      float* __restrict__ outW,
                      int* __restrict__ outI) {
    __shared__ unsigned long long keys[SEQ];

    const int be  = blockIdx.x;            // b * NUM_EXPERTS + e
    const int tid = threadIdx.x;
    const float* row = wsW + (size_t)be * SEQ;

    for (int i = tid; i < SEQ; i += 512) {
        unsigned int bits = __float_as_uint(row[i]);   // positive floats: monotonic
        keys[i] = ((unsigned long long)bits << 32) | (0xFFFFFFFFu - (unsigned)i);
    }

    for (int k = 2; k <= SEQ; k <<= 1) {
        for (int jj = k >> 1; jj > 0; jj >>= 1) {
            __syncthreads();
            for (int t = tid; t < SEQ / 2; t += 512) {
                int pos = ((t & ~(jj - 1)) << 1) | (t & (jj - 1));
                int partner = pos | jj;
                bool desc = ((pos & k) == 0);
                unsigned long long a = keys[pos];
                unsigned long long b = keys[partner];
                bool doswap = desc ? (a < b) : (a > b);
                if (doswap) { keys[pos] = b; keys[partner] = a; }
            }
        }
    }
    __syncthreads();

    const int bI = be >> 3;       // batch
    const int eI = be & 7;        // expert
    for (int cpos = tid; cpos < CAPACITY; cpos += 512) {
        unsigned long long kk = keys[cpos];
        float wv = __uint_as_float((unsigned int)(kk >> 32));
        int idx  = (int)(0xFFFFFFFFu - (unsigned int)(kk & 0xFFFFFFFFull));
        size_t o = ((size_t)bI * CAPACITY + cpos) * NUM_EXPERTS + eI;
        outW[o] = wv;
        outI[o] = idx;
    }
}

extern "C" void kernel_launch(void* const* d_in, const int* in_sizes, int n_in,
                              void* d_out, int out_size, void* d_ws, size_t ws_size,
                              hipStream_t stream) {
    const float* hidden = (const float*)d_in[0];   // (4, 4096, 1024) f32
    const float* W      = (const float*)d_in[1];   // (8, 1024) f32
    const float* bias   = (const float*)d_in[2];   // (8,) f32

    float* wsW = (float*)d_ws;                     // (B, E, S) = 512 KB scratch

    float* outW = (float*)d_out;                              // (4, 640, 8) f32
    int*   outI = (int*)((float*)d_out + BATCH * CAPACITY * NUM_EXPERTS);

    // Pass 1: logits (WMMA) + softmax -> transposed weights in scratch.
    router_softmax_kernel<<<(BATCH * SEQ) / 128, 256, 0, stream>>>(hidden, W, bias, wsW);
    // Pass 2: per-(b,e) bitonic top-640 with indices.
    topk_sort_kernel<<<BATCH * NUM_EXPERTS, 512, 0, stream>>>(wsW, outW, outI);
}